// LocalAttention2d_74637941670330
// MI455X (gfx1250) — compile-verified
//
#include <hip/hip_runtime.h>
#include <math.h>

typedef float v2f __attribute__((ext_vector_type(2)));
typedef float v8f __attribute__((ext_vector_type(8)));

#define B_    8
#define CQ    256
#define H_    128
#define Wd_   128
#define T_    1024
#define CC    256
#define P_    128
#define ROWS  (H_ + 1)   // 129 (padded)
#define COLS  (Wd_ + 1)  // 129 (padded)
#define TILE  16         // t's per workgroup

// One workgroup = 16 consecutive t of one batch b. 256 threads = 8 wave32.
__global__ __launch_bounds__(256)
void local_attn2d_kernel(const float* __restrict__ q,    // [B,CQ,H,W]
                         const float* __restrict__ ct,   // [B,T,CC]
                         const float* __restrict__ Wa,   // [CC,CQ]
                         const float* __restrict__ Wp,   // [P,CC]
                         const float* __restrict__ Vp,   // [2,P]
                         float* __restrict__ out)        // [B,T,CQ]
{
    __shared__ float s_ct[TILE][CC + 1];   // c_t tile (padded: bank-conflict-free A loads)
    __shared__ float s_wc[TILE][CQ + 1];   // Wc tile
    __shared__ float s_h [TILE][P_ + 1];   // tanh(ct @ Wp^T)
    __shared__ float s_p [TILE][2];        // p_t
    __shared__ int   s_ri[7], s_ci[7];
    __shared__ float s_rexp[7], s_cexp[7];
    __shared__ float s_scores[49];
    __shared__ float s_w[49];

    const int tid  = threadIdx.x;
    const int lane = tid & 31;
    const int wv   = tid >> 5;           // wave id 0..7
    const int b    = blockIdx.x / (T_ / TILE);
    const int t0   = (blockIdx.x % (T_ / TILE)) * TILE;

    // ---- load c_t tile [16 x 256] (coalesced) ----
    for (int i = 0; i < TILE; ++i)
        s_ct[i][tid] = ct[((size_t)(b * T_ + t0 + i)) * CC + tid];
    __syncthreads();

    // WMMA f32 16x16x4 fragment coordinates (ISA 7.12.2 layouts)
    const int mrow = lane & 15;          // A row
    const int klo  = (lane >> 4) << 1;   // lanes 0-15 -> K=0,1 ; 16-31 -> K=2,3
    const int row0 = (lane >> 4) << 3;   // C/D: M = vgpr + 8*(lane>=16)
    const int ncol = lane & 15;          // B/C/D column within 16-wide tile

    // ---- GEMM 1: hidden = c_t @ Wp^T  (M=16, N=128, K=256), tanh into LDS ----
    {
        const int n = wv * 16 + ncol;
        v8f acc = {};
        #pragma unroll 8
        for (int k0 = 0; k0 < CC; k0 += 4) {
            v2f a, bb;
            a[0]  = s_ct[mrow][k0 + klo];
            a[1]  = s_ct[mrow][k0 + klo + 1];
            bb[0] = Wp[(size_t)n * CC + k0 + klo];      // B[k][n] = Wp[n][k]
            bb[1] = Wp[(size_t)n * CC + k0 + klo + 1];
            acc = __builtin_amdgcn_wmma_f32_16x16x4_f32(
                      false, a, false, bb, (short)0, acc, false, false);
        }
        #pragma unroll
        for (int v = 0; v < 8; ++v)
            s_h[row0 + v][n] = tanhf(acc[v]);
    }

    // ---- GEMM 2: Wc = c_t @ Wa  (M=16, N=256, K=256) ----
    #pragma unroll
    for (int nt = 0; nt < 2; ++nt) {
        const int n = (wv + nt * 8) * 16 + ncol;
        v8f acc = {};
        #pragma unroll 8
        for (int k0 = 0; k0 < CC; k0 += 4) {
            v2f a, bb;
            a[0]  = s_ct[mrow][k0 + klo];
            a[1]  = s_ct[mrow][k0 + klo + 1];
            bb[0] = Wa[(size_t)(k0 + klo) * CQ + n];     // B[k][n] = Wa[k][n]
            bb[1] = Wa[(size_t)(k0 + klo + 1) * CQ + n];
            acc = __builtin_amdgcn_wmma_f32_16x16x4_f32(
                      false, a, false, bb, (short)0, acc, false, false);
        }
        #pragma unroll
        for (int v = 0; v < 8; ++v)
            s_wc[row0 + v][n] = acc[v];
    }
    __syncthreads();

    // ---- p_t = 128 * sigmoid(hidden @ Vp^T)  (32 threads: 16 t x 2 dims) ----
    if (tid < 32) {
        const int i = tid >> 1, d = tid & 1;
        float s = 0.f;
        #pragma unroll 4
        for (int p = 0; p < P_; ++p)
            s += s_h[i][p] * Vp[d * P_ + p];
        s_p[i][d] = (float)H_ / (1.f + expf(-s));
    }
    __syncthreads();

    // ---- per-t attention ----
    for (int i = 0; i < TILE; ++i) {
        const float p0 = s_p[i][0];
        const float p1 = s_p[i][1];

        // window indices: r = clip(round(p)+off+1, 0, 129) % 129  (0 => NaN pad slot)
        if (tid < 7) {
            int rr = __float2int_rn(p0) + (tid - 3) + 1;   // round-half-even == jnp.round
            rr = min(max(rr, 0), ROWS) % ROWS;
            s_ri[tid] = rr;
            float d = ((float)max(rr - 1, 0) - p0) * (1.0f / 3.0f);
            s_rexp[tid] = -2.f * d * d;
        } else if (tid >= 8 && tid < 15) {
            const int j = tid - 8;
            int cc = __float2int_rn(p1) + (j - 3) + 1;
            cc = min(max(cc, 0), COLS) % COLS;
            s_ci[j] = cc;
            float d = ((float)max(cc - 1, 0) - p1) * (1.0f / 3.0f);
            s_cexp[j] = -2.f * d * d;
        }
        __syncthreads();

        // scores[k] = gathered[k,:] . Wc  ; 49 slots x 4 channel-partials
        if (tid < 196) {
            const int kk = tid >> 2, g = tid & 3;
            const int ii = kk / 7, jj = kk % 7;
            const int r = s_ri[ii], c = s_ci[jj];
            if (r != 0 && c != 0) {
                const float* qp = q + (size_t)b * CQ * H_ * Wd_
                                    + (size_t)(r - 1) * Wd_ + (c - 1);
                float partial = 0.f;
                #pragma unroll 4
                for (int ch = g; ch < CQ; ch += 4)
                    partial += qp[(size_t)ch * (H_ * Wd_)] * s_wc[i][ch];
                partial += __shfl_xor(partial, 1, 4);
                partial += __shfl_xor(partial, 2, 4);
                if (g == 0) s_scores[kk] = partial;
            } else if (g == 0) {
                s_scores[kk] = -__builtin_inff();
            }
        }
        __syncthreads();

        // masked softmax fused with Gaussian decay (tiny: one thread)
        if (tid == 0) {
            float m = -__builtin_inff();
            for (int k = 0; k < 49; ++k) m = fmaxf(m, s_scores[k]);
            float sum = 0.f;
            for (int k = 0; k < 49; ++k) {
                float e = expf(s_scores[k] - m);      // masked -> 0
                sum += e;
                s_w[k] = e * expf(s_rexp[k / 7] + s_cexp[k % 7]);
            }
            const float inv = 1.f / sum;
            for (int k = 0; k < 49; ++k) s_w[k] *= inv;
        }
        __syncthreads();

        // out[ch] = sum_k w[k] * gathered[k][ch]  (thread == channel)
        {
            float acc = 0.f;
            const size_t qbase = ((size_t)b * CQ + tid) * (size_t)(H_ * Wd_);
            #pragma unroll
            for (int ii = 0; ii < 7; ++ii) {
                const int r = s_ri[ii];
                if (r == 0) continue;
                #pragma unroll
                for (int jj = 0; jj < 7; ++jj) {
                    const int c = s_ci[jj];
                    if (c == 0) continue;
                    acc += s_w[ii * 7 + jj] *
                           q[qbase + (size_t)(r - 1) * Wd_ + (c - 1)];
                }
            }
            out[((size_t)(b * T_ + t0 + i)) * CQ + tid] = acc;
        }
        __syncthreads();
    }
}

extern "C" void kernel_launch(void* const* d_in, const int* in_sizes, int n_in,
                              void* d_out, int out_size, void* d_ws, size_t ws_size,
                              hipStream_t stream) {
    (void)in_sizes; (void)n_in; (void)out_size; (void)d_ws; (void)ws_size;
    const float* q   = (const float*)d_in[0];
    const float* ct  = (const float*)d_in[1];
    const float* Wa  = (const float*)d_in[2];
    const float* Wp  = (const float*)d_in[3];
    const float* Vp  = (const float*)d_in[4];
    float* out = (float*)d_out;

    dim3 grid(B_ * (T_ / TILE));   // 512 workgroups
    dim3 block(256);               // 8 wave32
    local_attn2d_kernel<<<grid, block, 0, stream>>>(q, ct, Wa, Wp, Vp, out);
}